// NeuralODE_22720376996221
// MI455X (gfx1250) — compile-verified
//
#include <hip/hip_runtime.h>
#include <math.h>

// Problem constants (from reference): B=131072, SEQ=150, D=2, H=64, PRED=150
#define B_TOT 131072
#define SEQQ  150
#define DD    2
#define HH    64
#define PREDN 150

typedef float v2f __attribute__((ext_vector_type(2)));
typedef float v8f __attribute__((ext_vector_type(8)));

__device__ __forceinline__ float dev_tanh(float v) {
#if __has_builtin(__builtin_amdgcn_tanhf)
    return __builtin_amdgcn_tanhf(v);          // gfx1250 v_tanh_f32
#elif __has_builtin(__builtin_amdgcn_tanh_f32)
    return __builtin_amdgcn_tanh_f32(v);
#else
    return tanhf(v);
#endif
}

// Swap the two 16-lane halves of a wave32 (VALU permlane; no LDS round-trip).
__device__ __forceinline__ float half_swap(float v) {
#if __has_builtin(__builtin_amdgcn_permlanex16)
    int r = __builtin_amdgcn_permlanex16(__float_as_int(v), __float_as_int(v),
                                         0x76543210, 0xfedcba98, false, false);
    return __int_as_float(r);
#else
    return __shfl_xor(v, 16, 32);
#endif
}

// One wave integrates 16 batch elements. State held transposed:
// s0,s1 are per-lane scalars; lane L holds batch (waveBase + L%16).
// Both lane halves carry identical state so EXEC stays all-ones for WMMA.
__global__ __launch_bounds__(256) void node_rk4_wmma_kernel(
    const float* __restrict__ x,   // (B, SEQ, D)
    const float* __restrict__ W1,  // (D, H) row-major: W1[d*H + h]
    const float* __restrict__ b1,  // (H)
    const float* __restrict__ W2,  // (H, D) row-major: W2[h*D + d]
    const float* __restrict__ b2,  // (D)
    float* __restrict__ out)       // (B, PRED, D)
{
    const int lane  = threadIdx.x & 31;
    const int wave  = (blockIdx.x * blockDim.x + threadIdx.x) >> 5;
    const int l16   = lane & 15;
    const bool hi   = lane >= 16;
    const int batch = wave * 16 + l16;     // both halves mirror the same batch

    // ---- Loop-invariant fragments -------------------------------------
    // A = W1^T tile (16x4, f32). Documented layout:
    //   comp0: K=0 (lanes0-15) / K=2 (lanes16-31); comp1: K=1 / K=3.
    // K=2,3 are zero-padded -> upper-lane B garbage contributes exactly 0.
    v2f afrag[4];
    // C bias fragments (16x16 f32 C layout: VGPR v -> row M=v (lo) / v+8 (hi))
    v8f cb1[4];
    // W2 coefficients in the same C layout for the VALU second GEMM
    v8f w2f0[4], w2f1[4];
#pragma unroll
    for (int t = 0; t < 4; ++t) {
        const int m = t * 16 + l16;
        afrag[t].x = hi ? 0.0f : W1[0 * HH + m];
        afrag[t].y = hi ? 0.0f : W1[1 * HH + m];
#pragma unroll
        for (int v = 0; v < 8; ++v) {
            const int row = t * 16 + v + (hi ? 8 : 0);   // hidden index of slot
            cb1[t][v]  = b1[row];
            w2f0[t][v] = W2[row * DD + 0];
            w2f1[t][v] = W2[row * DD + 1];
        }
    }
    const float bb2_0 = b2[0];
    const float bb2_1 = b2[1];

    // ---- Initial state: y0 = x[b, SEQ-1, :] ---------------------------
    const size_t xoff = (size_t)batch * (SEQQ * DD) + (size_t)(SEQQ - 1) * DD;
    float s0 = x[xoff + 0];
    float s1 = x[xoff + 1];

    float* orow = out + (size_t)batch * (PREDN * DD);
    if (!hi) {
        orow[0] = s0;
        orow[1] = s1;
    }

    // f(y) = tanh(y@W1 + b1) @ W2 + b2, evaluated for 16 states at once.
    auto feval = [&](float y0v, float y1v, float& o0, float& o1) {
        v2f bfrag;
        bfrag.x = hi ? y1v : y0v;   // row K=0 (lo); correct under either B layout
        bfrag.y = y1v;              // row K=1 (lo); rows K=2,3 hit zero A columns

        // Issue all 4 matrix ops first: Hpre^T tiles = W1^T @ Y^T + b1.
        v8f d[4];
#pragma unroll
        for (int t = 0; t < 4; ++t)
            d[t] = __builtin_amdgcn_wmma_f32_16x16x4_f32(
                false, afrag[t], false, bfrag, (short)0, cb1[t], false, false);

        // 32 independent tanh back-to-back: keeps the TRANS pipe pipelined
        // (each tanh is the "independent op" covering the previous one's
        // hazard window -> no v_nop filler).
#pragma unroll
        for (int t = 0; t < 4; ++t)
#pragma unroll
            for (int v = 0; v < 8; ++v)
                d[t][v] = dev_tanh(d[t][v]);

        // Second GEMM (K=64, N=2) in VALU with two accumulator pairs so
        // consecutive dual-FMAs don't chain on the same destinations.
        float a0e = 0.0f, a1e = 0.0f, a0o = 0.0f, a1o = 0.0f;
#pragma unroll
        for (int t = 0; t < 4; ++t) {
#pragma unroll
            for (int v = 0; v < 8; v += 2) {
                a0e = fmaf(d[t][v],     w2f0[t][v],     a0e);
                a1e = fmaf(d[t][v],     w2f1[t][v],     a1e);
                a0o = fmaf(d[t][v + 1], w2f0[t][v + 1], a0o);
                a1o = fmaf(d[t][v + 1], w2f1[t][v + 1], a1o);
            }
        }
        float a0 = a0e + a0o;
        float a1 = a1e + a1o;
        // Fold lane halves so ALL lanes hold the full 64-term sum.
        a0 += half_swap(a0);
        a1 += half_swap(a1);
        o0 = a0 + bb2_0;
        o1 = a1 + bb2_1;
    };

    const float tstep = 150.0f / 149.0f;   // jnp.linspace(0,150,150) spacing

#pragma clang loop unroll(disable)
    for (int i = 0; i < PREDN - 1; ++i) {
        const float dt = (float)(i + 1) * tstep - (float)i * tstep;

        float k1_0, k1_1, k2_0, k2_1, k3_0, k3_1, k4_0, k4_1;
        feval(s0, s1, k1_0, k1_1);
        feval(fmaf(dt * (1.0f / 3.0f), k1_0, s0),
              fmaf(dt * (1.0f / 3.0f), k1_1, s1), k2_0, k2_1);
        feval(fmaf(dt, k2_0 - k1_0 * (1.0f / 3.0f), s0),
              fmaf(dt, k2_1 - k1_1 * (1.0f / 3.0f), s1), k3_0, k3_1);
        feval(fmaf(dt, k1_0 - k2_0 + k3_0, s0),
              fmaf(dt, k1_1 - k2_1 + k3_1, s1), k4_0, k4_1);

        s0 = fmaf(dt * 0.125f, k1_0 + 3.0f * (k2_0 + k3_0) + k4_0, s0);
        s1 = fmaf(dt * 0.125f, k1_1 + 3.0f * (k2_1 + k3_1) + k4_1, s1);

        if (!hi) {
            float2* o = (float2*)(orow + (size_t)(i + 1) * DD);
            *o = make_float2(s0, s1);
        }
    }
}

extern "C" void kernel_launch(void* const* d_in, const int* in_sizes, int n_in,
                              void* d_out, int out_size, void* d_ws, size_t ws_size,
                              hipStream_t stream) {
    const float* x  = (const float*)d_in[0];
    const float* W1 = (const float*)d_in[1];
    const float* b1 = (const float*)d_in[2];
    const float* W2 = (const float*)d_in[3];
    const float* b2 = (const float*)d_in[4];
    float* out = (float*)d_out;

    // 16 batches per wave, 8 waves per 256-thread block -> 128 batches/block
    const int blocks = B_TOT / 128;   // 1024
    node_rk4_wmma_kernel<<<blocks, 256, 0, stream>>>(x, W1, b1, W2, b2, out);
}